// PartialProductAccumulator_84713934946554
// MI455X (gfx1250) — compile-verified
//
#include <hip/hip_runtime.h>

// Problem constants (from the reference)
#define N_PP    54
#define BATCH   16384
#define BITS    108
#define S_ELEMS (BATCH * BITS)          // 1,769,472 floats per partial-product slice

// Kernel A config: column sums. Two float4 (8 bit-columns) per thread per stage.
#define A_BLOCK 256
#define A_VEC4  2                                    // float4 elements per thread
#define A_TILE  (A_BLOCK * A_VEC4)                   // 512 float4 per block per slice
#define A_GRID  (S_ELEMS / (4 * A_TILE))             // 864 blocks, exact
#define NBUF    4                                    // async pipeline depth

// Kernel B config: carry propagation. One batch row per thread.
#define B_ROWS  128
#define B_GRID  (BATCH / B_ROWS)            // 128 blocks, exact
#define UPITCH  29                           // 27 u32 words/row, gcd(29,64)=1 -> conflict-free
#define FPITCH  109                          // 108 floats/row, gcd(45,64)=1 -> conflict-free

static_assert(S_ELEMS % (4 * A_TILE) == 0, "grid A exact");
static_assert(BATCH % B_ROWS == 0, "grid B exact");
static_assert(BITS % 4 == 0, "bits pack into u32");

// ---- CDNA5 split-counter waits (inline asm with memory clobber so the
// ---- compiler cannot hoist the LDS reads above the wait). gfx1250 SOPP.
#define WAIT_ASYNC(n) asm volatile("s_wait_asynccnt %0" :: "n"(n) : "memory")
#define WAIT_DS(n)    asm volatile("s_wait_dscnt %0"    :: "n"(n) : "memory")

// Builtin signature (from hipcc): (<4 x i32> addrspace(1)*, <4 x i32> addrspace(3)*, imm, imm)
typedef int v4i __attribute__((ext_vector_type(4)));
typedef __attribute__((address_space(1))) v4i gv4i;
typedef __attribute__((address_space(3))) v4i lv4i;

// ---- Async global -> LDS 16B/lane copy (ASYNCcnt-tracked).
__device__ __forceinline__ void async_load_b128(const float* g, void* lds) {
#if __has_builtin(__builtin_amdgcn_global_load_async_to_lds_b128)
    __builtin_amdgcn_global_load_async_to_lds_b128(
        (gv4i*)const_cast<float*>(g), (lv4i*)lds, /*imm offset*/0, /*cpol*/0);
#else
    unsigned lds_addr =
        (unsigned)(unsigned long long)(__attribute__((address_space(3))) void*)lds;
    asm volatile("global_load_async_to_lds_b128 %0, %1, off"
                 :: "v"(lds_addr), "v"(g) : "memory");
#endif
}

// ===========================================================================
// Kernel A: s[b,j] = sum_i pps[i,b,j]; 54-deep streaming reduction through a
// 4-deep async-DMA LDS pipeline (6 in-flight b128 ops/wave steady-state).
// Packs 4 byte-sized sums (<=54) into one u32.
// ===========================================================================
__global__ void __launch_bounds__(A_BLOCK)
colsum_kernel(const float* __restrict__ pps, unsigned* __restrict__ sums) {
    __shared__ float4 stage[NBUF][A_TILE];

    const int tid          = threadIdx.x;
    const unsigned g4base  = blockIdx.x * A_TILE;       // block's first float4 in a slice
    const unsigned e0      = g4base + tid;              // this thread's two float4 slots
    const unsigned e1      = e0 + A_BLOCK;
    const float*   p0      = pps + (size_t)e0 * 4;
    const float*   p1      = pps + (size_t)e1 * 4;

    // Prologue: stage pps[0..2] into buffers 0..2 (6 async ops in flight).
#pragma unroll
    for (int s = 0; s < NBUF - 1; ++s) {
        async_load_b128(p0 + (size_t)s * S_ELEMS, &stage[s][tid]);
        async_load_b128(p1 + (size_t)s * S_ELEMS, &stage[s][tid + A_BLOCK]);
    }

    float4 acc0 = make_float4(0.f, 0.f, 0.f, 0.f);
    float4 acc1 = make_float4(0.f, 0.f, 0.f, 0.f);

    for (int i = 0; i < N_PP; ++i) {
        const int cur = i & (NBUF - 1);
        if (i + NBUF - 1 < N_PP) {
            // Buffer being refilled was consumed at iteration i-1; make sure
            // those ds_loads drained before the DMA overwrites it.
            WAIT_DS(0);
            const int nxt = (i + NBUF - 1) & (NBUF - 1);
            async_load_b128(p0 + (size_t)(i + NBUF - 1) * S_ELEMS, &stage[nxt][tid]);
            async_load_b128(p1 + (size_t)(i + NBUF - 1) * S_ELEMS, &stage[nxt][tid + A_BLOCK]);
            WAIT_ASYNC(6);   // in-order: <=6 outstanding => stage i complete
        } else {
            WAIT_ASYNC(0);   // tail: drain everything once
        }
        float4 v0 = stage[cur][tid];             // ds_load_b128
        float4 v1 = stage[cur][tid + A_BLOCK];   // ds_load_b128
        acc0.x += v0.x; acc0.y += v0.y; acc0.z += v0.z; acc0.w += v0.w;
        acc1.x += v1.x; acc1.y += v1.y; acc1.z += v1.z; acc1.w += v1.w;
    }

    // Sums are exact small integers in [0,54]; pack 4 bytes per u32.
    unsigned r0 = ((unsigned)(int)acc0.x)        | ((unsigned)(int)acc0.y << 8)
                | ((unsigned)(int)acc0.z << 16)  | ((unsigned)(int)acc0.w << 24);
    unsigned r1 = ((unsigned)(int)acc1.x)        | ((unsigned)(int)acc1.y << 8)
                | ((unsigned)(int)acc1.z << 16)  | ((unsigned)(int)acc1.w << 24);
    sums[e0] = r0;
    sums[e1] = r1;
}

// ===========================================================================
// Kernel B: per batch row, ripple the column sums:
//   t = s[j] + c; out[j] = t & 1; c = t >> 1   (final carry dropped == mod 2^108)
// LDS-staged on both sides so all global traffic is coalesced.
// ===========================================================================
__global__ void __launch_bounds__(B_ROWS)
carry_kernel(const unsigned* __restrict__ sums, float* __restrict__ out) {
    __shared__ unsigned utile[B_ROWS * UPITCH];
    __shared__ float    ftile[B_ROWS * FPITCH];

    const int tid            = threadIdx.x;
    const unsigned rowBase   = blockIdx.x * B_ROWS;
    const unsigned gUintBase = rowBase * (BITS / 4);   // 27 u32 words per row

    // Coalesced load of 128 rows x 27 words of packed byte-sums.
    for (int k = tid; k < B_ROWS * (BITS / 4); k += B_ROWS) {
        int b = k / (BITS / 4), w = k % (BITS / 4);
        utile[b * UPITCH + w] = sums[gUintBase + k];
    }
    __syncthreads();

    // Per-row ripple carry over 108 bit positions.
    unsigned c = 0;
#pragma unroll
    for (int w = 0; w < BITS / 4; ++w) {
        unsigned u = utile[tid * UPITCH + w];
#pragma unroll
        for (int k = 0; k < 4; ++k) {
            unsigned s = (u >> (8 * k)) & 0xFFu;
            unsigned t = s + c;
            ftile[tid * FPITCH + w * 4 + k] = (float)(t & 1u);
            c = t >> 1;
        }
    }
    __syncthreads();

    // Coalesced store of 128 rows x 108 output bits (as floats).
    const unsigned gOutBase = rowBase * BITS;
    for (int k = tid; k < B_ROWS * BITS; k += B_ROWS) {
        int b = k / BITS, j = k % BITS;
        out[gOutBase + k] = ftile[b * FPITCH + j];
    }
}

extern "C" void kernel_launch(void* const* d_in, const int* in_sizes, int n_in,
                              void* d_out, int out_size, void* d_ws, size_t ws_size,
                              hipStream_t stream) {
    const float* pps  = (const float*)d_in[0];          // [54, 16384, 108] f32 in {0,1}
    unsigned*    sums = (unsigned*)d_ws;                // 442,368 u32 = 1.73 MB scratch
    float*       out  = (float*)d_out;                  // [16384, 108] f32

    colsum_kernel<<<A_GRID, A_BLOCK, 0, stream>>>(pps, sums);
    carry_kernel<<<B_GRID, B_ROWS, 0, stream>>>(sums, out);
}